// StaticSceneFeatureExtractorGrid_73392401154335
// MI455X (gfx1250) — compile-verified
//
#include <hip/hip_runtime.h>
#include <hip/hip_bf16.h>
#include <math.h>

#define H_DIM   64
#define ATT_DIM 32
#define GCELLS  64   // 8x8 grid

typedef float v2f __attribute__((ext_vector_type(2)));
typedef float v8f __attribute__((ext_vector_type(8)));

#ifndef __has_builtin
#define __has_builtin(x) 0
#endif
#if __has_builtin(__builtin_amdgcn_wmma_f32_16x16x4_f32)
#define USE_WMMA_F32 1
#else
#define USE_WMMA_F32 0
#endif

// ---------------------------------------------------------------------------
// Kernel 1: per-pedestrian 8x8 occupancy histogram.
// One wave (32 lanes) per pedestrian; 64-bin histogram in a private LDS slice
// (ds_add_u32, no cross-wave traffic). Scene (64KB) stays hot in L0/L2.
// ---------------------------------------------------------------------------
__global__ void occ_kernel(const float2* __restrict__ scene,
                           const float2* __restrict__ pos,
                           float* __restrict__ e, int num_pts) {
  __shared__ unsigned int hist[8][GCELLS];
  const int wave = threadIdx.x >> 5;
  const int lane = threadIdx.x & 31;
  const int n = blockIdx.x * 8 + wave;

  unsigned int* hb = hist[wave];
  hb[lane] = 0u;
  hb[lane + 32] = 0u;
  __syncthreads();

  const float2 p = pos[n];
  const float tlx = p.x - 1.0f, brx = p.x + 1.0f;   // NEIGHBORHOOD/2 = 1.0
  const float tly = p.y + 1.0f, bry = p.y - 1.0f;

  for (int i = lane; i < num_pts; i += 32) {
    const float2 sp = scene[i];
    // inside = !(sx>=brx | sx<=tlx | sy>=tly | sy<=bry)  (strict, matches ref)
    const bool inside = (sp.x < brx) && (sp.x > tlx) &&
                        (sp.y < tly) && (sp.y > bry);
    if (inside) {
      // ref: floor((sx-tlx)/2*8) == floor((sx-tlx)*4) exactly in f32
      const int cx = (int)floorf((sp.x - tlx) * 4.0f);
      const int cy = (int)floorf((tly - sp.y) * 4.0f);
      const int id = cx + (cy << 3);
      if (id >= 0 && id < GCELLS) atomicAdd(&hb[id], 1u);
    }
  }
  __syncthreads();
  e[n * GCELLS + lane]      = (float)hb[lane];
  e[n * GCELLS + lane + 32] = (float)hb[lane + 32];
}

// ---------------------------------------------------------------------------
// Kernel 2: C[M,N] = A[M,64] @ B[64,N], f32, exact, via V_WMMA_F32_16X16X4_F32.
// One wave per 16x16 C tile; 16 chained WMMAs over K=64.
// Fragment layouts per CDNA5 ISA 7.12.2:
//   A 16x4: lane<16 -> K={0,1}, lane>=16 -> K={2,3}; M = lane&15
//   B 4x16: lane<16 -> K={0,1}, lane>=16 -> K={2,3}; N = lane&15
//   C/D   : VGPR r -> M = r + 8*(lane>=16); N = lane&15
// ---------------------------------------------------------------------------
__global__ void gemm64_wmma(const float* __restrict__ A, const float* __restrict__ B,
                            float* __restrict__ C, int M, int N) {
  const int K = 64;
  const int lane = threadIdx.x & 31;
  const int waves_per_block = blockDim.x >> 5;
  const int tile = blockIdx.x * waves_per_block + (threadIdx.x >> 5);
  const int tilesN = N >> 4;
  const int totalTiles = (M >> 4) * tilesN;
  if (tile >= totalTiles) return;           // wave-uniform: EXEC stays all-1s

  const int tm = (tile / tilesN) << 4;
  const int tn = (tile % tilesN) << 4;
  const int half = lane >> 4;               // 0 or 1
  const int idx  = lane & 15;

  v8f acc = {0.f, 0.f, 0.f, 0.f, 0.f, 0.f, 0.f, 0.f};

#if USE_WMMA_F32
  for (int k = 0; k < K; k += 4) {
    const int kk = k + (half << 1);
    v2f a, b;
    a.x = A[(tm + idx) * K + kk];
    a.y = A[(tm + idx) * K + kk + 1];
    b.x = B[kk * N + tn + idx];
    b.y = B[(kk + 1) * N + tn + idx];
    acc = __builtin_amdgcn_wmma_f32_16x16x4_f32(
        /*neg_a=*/false, a, /*neg_b=*/false, b,
        /*c_mod=*/(short)0, acc, /*reuse_a=*/false, /*reuse_b=*/false);
  }
#else
  // scalar fallback computing the same lane-mapped outputs
  for (int r = 0; r < 8; ++r) {
    const int row = tm + r + (half << 3);
    float s = 0.f;
    for (int k = 0; k < K; ++k) s = fmaf(A[row * K + k], B[k * N + tn + idx], s);
    acc[r] = s;
  }
#endif

  for (int r = 0; r < 8; ++r) {
    const int row = tm + r + (half << 3);
    C[row * N + tn + idx] = acc[r];
  }
}

// ---------------------------------------------------------------------------
// Kernel 3: fused attention + softmax + LSTM cell. One wave per pedestrian.
// ---------------------------------------------------------------------------
__device__ __forceinline__ float wave_sum(float v) {
  for (int o = 16; o > 0; o >>= 1) v += __shfl_xor(v, o, 32);
  return v;
}
__device__ __forceinline__ float wave_max(float v) {
  for (int o = 16; o > 0; o >>= 1) v = fmaxf(v, __shfl_xor(v, o, 32));
  return v;
}
__device__ __forceinline__ float sigmoidf(float x) {
  return 1.f / (1.f + expf(-x));
}

__global__ void fused_attn_lstm(
    const float* __restrict__ e,     // [N,64] occupancy counts
    const float* __restrict__ att2,  // [N,32] h @ W_dec (no bias)
    const float* __restrict__ ghh,   // [N,256] h @ W_hh (no bias)
    const float* __restrict__ h,     // [N,64]
    const float* __restrict__ pos,   // [N,2]
    const float* __restrict__ disp,  // [N,2]
    const float* __restrict__ W_enc, const float* __restrict__ b_enc,
    const float* __restrict__ b_dec,
    const float* __restrict__ W_full, const float* __restrict__ b_full,
    const float* __restrict__ W_fb, const float* __restrict__ b_fb,
    const float* __restrict__ W_emb, const float* __restrict__ b_emb,
    const float* __restrict__ W_ih, const float* __restrict__ b_ih,
    const float* __restrict__ b_hh,
    float* __restrict__ out, int num_ped) {
  const int wave = threadIdx.x >> 5;
  const int lane = threadIdx.x & 31;
  const int n = blockIdx.x * (blockDim.x >> 5) + wave;
  if (n >= num_ped) return;

  // --- attention logits: lane owns cells {lane, lane+32} ---
  const float e0 = e[n * 64 + lane];
  const float e1 = e[n * 64 + lane + 32];
  const float bf = b_full[0];
  float a0 = bf, a1 = bf;
  for (int k = 0; k < ATT_DIM; ++k) {
    const float w = W_enc[k];                              // W_enc is [1,32]
    const float s = att2[n * ATT_DIM + k] + b_dec[k] + b_enc[k];
    const float f = W_full[k];                             // W_full is [32,1]
    a0 += fmaxf(fmaf(e0, w, s), 0.f) * f;
    a1 += fmaxf(fmaf(e1, w, s), 0.f) * f;
  }

  // --- softmax over 64 cells + weighted sum of e ---
  const float m  = wave_max(fmaxf(a0, a1));
  const float p0 = expf(a0 - m), p1 = expf(a1 - m);
  const float ssum = wave_sum(p0 + p1);
  const float wsum = wave_sum(e0 * p0 + e1 * p1);
  const float awe = wsum / ssum;

  // --- gate = sigmoid(h @ W_fbeta + b) ---
  const float hv0 = h[n * 64 + lane], hv1 = h[n * 64 + lane + 32];
  const float gd = wave_sum(hv0 * W_fb[lane] + hv1 * W_fb[lane + 32]);
  const float xa = sigmoidf(gd + b_fb[0]) * awe;

  // --- embedding: [px,py,dx,dy] @ W_emb[4,4] + b_emb ---
  const float px = pos[2 * n], py = pos[2 * n + 1];
  const float dx = disp[2 * n], dy = disp[2 * n + 1];
  float em[4];
  for (int j = 0; j < 4; ++j)
    em[j] = px * W_emb[j] + py * W_emb[4 + j] + dx * W_emb[8 + j] +
            dy * W_emb[12 + j] + b_emb[j];

  // --- LSTM gates; f-gate is dead (no prior cell state). d in {lane,lane+32}
  auto gv = [&](int mc) -> float {
    float s = ghh[n * 256 + mc] + b_ih[mc] + b_hh[mc];
    s = fmaf(em[0], W_ih[mc],        s);
    s = fmaf(em[1], W_ih[256 + mc],  s);
    s = fmaf(em[2], W_ih[512 + mc],  s);
    s = fmaf(em[3], W_ih[768 + mc],  s);
    s = fmaf(xa,    W_ih[1024 + mc], s);
    return s;
  };
  for (int t = 0; t < 2; ++t) {
    const int d = lane + t * 32;
    const float gi = gv(d);          // input gate
    const float gg = gv(128 + d);    // candidate
    const float go = gv(192 + d);    // output gate
    const float c = sigmoidf(gi) * tanhf(gg);
    out[n * 64 + d] = sigmoidf(go) * tanhf(c);
  }
}

// ---------------------------------------------------------------------------
extern "C" void kernel_launch(void* const* d_in, const int* in_sizes, int n_in,
                              void* d_out, int out_size, void* d_ws, size_t ws_size,
                              hipStream_t stream) {
  const float* scene  = (const float*)d_in[0];
  const float* pos    = (const float*)d_in[1];
  const float* disp   = (const float*)d_in[2];
  const float* h      = (const float*)d_in[3];
  const float* W_enc  = (const float*)d_in[4];
  const float* b_enc  = (const float*)d_in[5];
  const float* W_dec  = (const float*)d_in[6];
  const float* b_dec  = (const float*)d_in[7];
  const float* W_full = (const float*)d_in[8];
  const float* b_full = (const float*)d_in[9];
  const float* W_fb   = (const float*)d_in[10];
  const float* b_fb   = (const float*)d_in[11];
  const float* W_emb  = (const float*)d_in[12];
  const float* b_emb  = (const float*)d_in[13];
  const float* W_ih   = (const float*)d_in[14];
  const float* b_ih   = (const float*)d_in[15];
  const float* W_hh   = (const float*)d_in[16];
  const float* b_hh   = (const float*)d_in[17];

  const int num_ped = in_sizes[3] / H_DIM;   // 2048 (host-side; can't read device scalar in capture)
  const int num_pts = in_sizes[0] / 2;       // 8192

  float* out  = (float*)d_out;
  float* e    = (float*)d_ws;                              // [N,64]
  float* att2 = e    + (size_t)num_ped * GCELLS;           // [N,32]
  float* ghh  = att2 + (size_t)num_ped * ATT_DIM;          // [N,256]

  // 1) occupancy grid: 8 waves/block, one ped per wave
  occ_kernel<<<num_ped / 8, 256, 0, stream>>>(
      (const float2*)scene, (const float2*)pos, e, num_pts);

  // 2) att2 = h @ W_dec   (N=32)
  {
    const int tiles = (num_ped / 16) * (ATT_DIM / 16);
    gemm64_wmma<<<(tiles + 3) / 4, 128, 0, stream>>>(h, W_dec, att2, num_ped, ATT_DIM);
  }
  // 3) ghh = h @ W_hh     (N=256)
  {
    const int tiles = (num_ped / 16) * ((4 * H_DIM) / 16);
    gemm64_wmma<<<(tiles + 3) / 4, 128, 0, stream>>>(h, W_hh, ghh, num_ped, 4 * H_DIM);
  }
  // 4) fused attention + LSTM
  fused_attn_lstm<<<num_ped / 8, 256, 0, stream>>>(
      e, att2, ghh, h, pos, disp,
      W_enc, b_enc, b_dec, W_full, b_full, W_fb, b_fb,
      W_emb, b_emb, W_ih, b_ih, b_hh, out, num_ped);
}